// GCN_body_19911468384625
// MI455X (gfx1250) — compile-verified
//
#include <hip/hip_runtime.h>

typedef float v2f __attribute__((ext_vector_type(2)));
typedef float v8f __attribute__((ext_vector_type(8)));

#define F 128
#define LN_EPS 1e-5f

// ---------------- degree / normalization ----------------
__global__ void k_deg(const int* __restrict__ dst, int E, int* __restrict__ cnt) {
    int e = blockIdx.x * blockDim.x + threadIdx.x;
    if (e < E) atomicAdd(&cnt[dst[e]], 1);
}

__global__ void k_dinv(const int* __restrict__ cnt, int N, float* __restrict__ dinv) {
    int v = blockIdx.x * blockDim.x + threadIdx.x;
    if (v < N) dinv[v] = rsqrtf((float)(cnt[v] + 1));   // +1 = self loop
}

// ---------------- GEMM + fused self-loop epilogue ----------------
// H   = X @ W                      (kept for the edge gather)
// agg = H * dinv^2 + bias          (self-loop contribution + bias, scatter target)
// Wave-per-16-row-strip, full 128 cols per wave (8 accumulator tiles).
// W staged once per block in LDS in a K-paired layout:
//   sW[(k/2)*256 + col*2 + (k&1)] = W[k][col]
// so each WMMA B fragment {W[k][col], W[k+1][col]} is one contiguous
// 8-byte ds_load_b64 straight into the even-aligned operand pair.
__global__ __launch_bounds__(256) void k_gemm(const float* __restrict__ X,
                                              const float* __restrict__ W,
                                              const float* __restrict__ dinv,
                                              const float* __restrict__ bias,
                                              float* __restrict__ H,
                                              float* __restrict__ agg, int N) {
    __shared__ float sW[F * F];     // 64 KB
    for (int i = threadIdx.x; i < (F / 2) * (F / 4); i += 256) {  // 64 k-pairs x 32 col-groups
        int p = i >> 5;             // k-pair index (k = 2p, 2p+1)
        int c = (i & 31) * 4;       // column group
        float4 w0 = *(const float4*)&W[(2 * p)     * F + c];
        float4 w1 = *(const float4*)&W[(2 * p + 1) * F + c];
        v2f* d = (v2f*)&sW[p * 256 + c * 2];
        d[0] = (v2f){w0.x, w1.x};
        d[1] = (v2f){w0.y, w1.y};
        d[2] = (v2f){w0.z, w1.z};
        d[3] = (v2f){w0.w, w1.w};
    }
    __syncthreads();

    const int wave = threadIdx.x >> 5;
    const int lane = threadIdx.x & 31;
    const int tile = blockIdx.x * 8 + wave;        // 16-row tile index
    if (tile * 16 >= N) return;                    // wave-uniform: EXEC stays full

    const int col   = lane & 15;
    const int khalf = (lane >> 4) << 1;            // lanes 0-15 -> K+0/1, 16-31 -> K+2/3
    int row = tile * 16 + (lane & 15);
    if (row > N - 1) row = N - 1;                  // clamp (N is 16-aligned anyway)
    const float* xrow = X + (size_t)row * F;

    v8f acc[8];
    const v8f vzero = {0.f, 0.f, 0.f, 0.f, 0.f, 0.f, 0.f, 0.f};
#pragma unroll
    for (int j = 0; j < 8; ++j) acc[j] = vzero;

    for (int k0 = 0; k0 < F; k0 += 4) {
        v2f a;
        a.x = xrow[k0 + khalf];
        a.y = xrow[k0 + khalf + 1];
        // B fragments: one ds_load_b64 each, immediate offsets j*128B
        const v2f* bbase = (const v2f*)&sW[((k0 + khalf) >> 1) * 256 + col * 2];
#pragma unroll
        for (int j = 0; j < 8; ++j) {
            v2f b = bbase[j * 16];
            acc[j] = __builtin_amdgcn_wmma_f32_16x16x4_f32(
                false, a, false, b, (short)0, acc[j], false, false);
        }
    }

    // Fused epilogue.  C/D layout: VGPR i -> row (i + 8*(lane>=16)), col = lane&15
    const int rbase = tile * 16 + (lane >> 4) * 8;
    float nrm[8];
#pragma unroll
    for (int i = 0; i < 8; ++i) {
        int r = rbase + i;
        float di = dinv[r < N ? r : N - 1];
        nrm[i] = di * di;
    }
#pragma unroll
    for (int j = 0; j < 8; ++j) {
        float bj = bias[j * 16 + col];
#pragma unroll
        for (int i = 0; i < 8; ++i) {
            int r = rbase + i;
            if (r < N) {
                float h = acc[j][i];
                size_t o = (size_t)r * F + j * 16 + col;
                H[o]   = h;
                agg[o] = h * nrm[i] + bj;
            }
        }
    }
}

// ---------------- edge scatter: agg[d] += h[s] * dinv[s]*dinv[d] ----------------
// One wave32 per edge; lane l handles features 4l..4l+3 (coalesced float4 gather,
// hardware f32 atomics for the scatter).
__global__ __launch_bounds__(256) void k_scatter(const int* __restrict__ src,
                                                 const int* __restrict__ dst,
                                                 const float* __restrict__ dinv,
                                                 const float* __restrict__ H,
                                                 float* __restrict__ agg, int E) {
    int gw   = (blockIdx.x * blockDim.x + threadIdx.x) >> 5;
    int lane = threadIdx.x & 31;
    if (gw >= E) return;
    int s = src[gw], d = dst[gw];
    float nrm = dinv[s] * dinv[d];
    float4 hv = *(const float4*)&H[(size_t)s * F + lane * 4];
    float* o = &agg[(size_t)d * F + lane * 4];
    unsafeAtomicAdd(o + 0, hv.x * nrm);
    unsafeAtomicAdd(o + 1, hv.y * nrm);
    unsafeAtomicAdd(o + 2, hv.z * nrm);
    unsafeAtomicAdd(o + 3, hv.w * nrm);
}

// ---------------- out = LayerNorm(ReLU(agg)) ----------------
__global__ __launch_bounds__(256) void k_relu_ln(const float* __restrict__ agg,
                                                 const float* __restrict__ g,
                                                 const float* __restrict__ be,
                                                 float* __restrict__ out, int N) {
    int wave = threadIdx.x >> 5;
    int lane = threadIdx.x & 31;
    int row = blockIdx.x * 8 + wave;
    if (row >= N) return;

    float4 v = *(const float4*)&agg[(size_t)row * F + lane * 4];
    v.x = fmaxf(v.x, 0.f); v.y = fmaxf(v.y, 0.f);
    v.z = fmaxf(v.z, 0.f); v.w = fmaxf(v.w, 0.f);

    float s = v.x + v.y + v.z + v.w;
#pragma unroll
    for (int m = 16; m >= 1; m >>= 1) s += __shfl_xor(s, m, 32);
    float mu = s * (1.f / F);

    float dx = v.x - mu, dy = v.y - mu, dz = v.z - mu, dw = v.w - mu;
    float q = dx * dx + dy * dy + dz * dz + dw * dw;
#pragma unroll
    for (int m = 16; m >= 1; m >>= 1) q += __shfl_xor(q, m, 32);
    float rs = rsqrtf(q * (1.f / F) + LN_EPS);

    float4 gg = *(const float4*)&g[lane * 4];
    float4 bb = *(const float4*)&be[lane * 4];
    float4 o;
    o.x = dx * rs * gg.x + bb.x;
    o.y = dy * rs * gg.y + bb.y;
    o.z = dz * rs * gg.z + bb.z;
    o.w = dw * rs * gg.w + bb.w;
    *(float4*)&out[(size_t)row * F + lane * 4] = o;
}

// ---------------- driver ----------------
static inline size_t align256(size_t x) { return (x + 255) & ~(size_t)255; }

extern "C" void kernel_launch(void* const* d_in, const int* in_sizes, int n_in,
                              void* d_out, int out_size, void* d_ws, size_t ws_size,
                              hipStream_t stream) {
    const float* x   = (const float*)d_in[0];
    const int*   ei  = (const int*)  d_in[1];
    const float* W1  = (const float*)d_in[2];
    const float* b1  = (const float*)d_in[3];
    const float* W2  = (const float*)d_in[4];
    const float* b2  = (const float*)d_in[5];
    const float* g1  = (const float*)d_in[6];
    const float* be1 = (const float*)d_in[7];
    const float* g2  = (const float*)d_in[8];
    const float* be2 = (const float*)d_in[9];

    const int N = in_sizes[0] / F;
    const int E = in_sizes[1] / 2;
    const int* src = ei;
    const int* dst = ei + E;
    float* out = (float*)d_out;

    // workspace layout
    char* ws = (char*)d_ws;
    size_t off = 0;
    float* dinv = (float*)(ws + off); off = align256(off + (size_t)N * 4);
    int*   cnt  = (int*)  (ws + off); off = align256(off + (size_t)N * 4);
    float* H    = (float*)(ws + off); off = align256(off + (size_t)N * F * 4);
    float* agg  = (float*)(ws + off); off = align256(off + (size_t)N * F * 4);
    (void)ws_size; (void)n_in; (void)out_size;

    const int T = 256;
    dim3 blk(T);

    // degrees (shared by both layers)
    hipMemsetAsync(cnt, 0, (size_t)N * 4, stream);
    k_deg <<<(E + T - 1) / T, blk, 0, stream>>>(dst, E, cnt);
    k_dinv<<<(N + T - 1) / T, blk, 0, stream>>>(cnt, N, dinv);

    const int gGemm = (N + 127) / 128;             // 8 waves x 16 rows per block
    const int gScat = (E + 7) / 8;                 // 8 edge-waves per block
    const int gLN   = (N + 7) / 8;                 // 8 row-waves per block

    // ---- layer 1 ----
    k_gemm   <<<gGemm, blk, 0, stream>>>(x, W1, dinv, b1, H, agg, N);
    k_scatter<<<gScat, blk, 0, stream>>>(src, dst, dinv, H, agg, E);
    k_relu_ln<<<gLN,   blk, 0, stream>>>(agg, g1, be1, out, N);

    // ---- layer 2 (input = layer-1 output in d_out) ----
    k_gemm   <<<gGemm, blk, 0, stream>>>(out, W2, dinv, b2, H, agg, N);
    k_scatter<<<gScat, blk, 0, stream>>>(src, dst, dinv, H, agg, E);
    k_relu_ln<<<gLN,   blk, 0, stream>>>(agg, g2, be2, out, N);
}